// LinearLLM_40810779246808
// MI455X (gfx1250) — compile-verified
//
#include <hip/hip_runtime.h>

typedef __attribute__((ext_vector_type(2))) float v2f;
typedef __attribute__((ext_vector_type(8))) float v8f;

#define S_LEN   513
#define V_TOK   5
#define W_DIM   32
#define B_SZ    16
#define N_COLS  (S_LEN * V_TOK)          // 2565
#define K_LEN   (S_LEN * W_DIM)          // 16416
#define K4_STEPS (K_LEN / 4)             // 4104
#define N_TILES ((N_COLS + 15) / 16)     // 161
#define WAVES_PER_BLK 16
#define BLK_THREADS (WAVES_PER_BLK * 32) // 512

// ---------------------------------------------------------------------------
// Kernel 1: embedding gather + pack x into WMMA 16x16x4 A-fragment layout.
// xpack layout: [k4][lane] as float2, lane<16 -> M=lane, K={0,1}of step;
// lane>=16 -> M=lane-16, K={2,3}. Main kernel then does coalesced b64 loads.
// ---------------------------------------------------------------------------
__global__ void pack_A_kernel(const int* __restrict__ src,
                              const float* __restrict__ emb,
                              float* __restrict__ xpack) {
    int idx = blockIdx.x * blockDim.x + threadIdx.x;   // one float2 per thread
    if (idx >= K4_STEPS * 32) return;
    int k4   = idx >> 5;
    int lane = idx & 31;
    int m    = lane & 15;
    int koff = (lane >> 4) << 1;
    int kb   = k4 * 4 + koff;

    v2f o;
    {
        int k = kb, s = k >> 5, w = k & 31;
        o.x = emb[src[m * S_LEN + s] * W_DIM + w];
    }
    {
        int k = kb + 1, s = k >> 5, w = k & 31;
        o.y = emb[src[m * S_LEN + s] * W_DIM + w];
    }
    ((v2f*)xpack)[idx] = o;
}

// ---------------------------------------------------------------------------
// Kernel 2: causal-masked skinny GEMM  C(16 x 2565) = A(16 x 16416) * W.
// weight (S,W,T,V) row-major == B-matrix (K=16416, N=2565) row-major.
// One workgroup per 16-column N-tile; 16 waves split the K (s-block) range.
// Causal mask applied branchlessly per lane so EXEC stays all-ones for WMMA.
// ---------------------------------------------------------------------------
__global__ void __launch_bounds__(BLK_THREADS)
masked_gemm_kernel(const float* __restrict__ xpack,
                   const float* __restrict__ weight,
                   const float* __restrict__ bias,
                   float* __restrict__ out) {
    __shared__ float lds[WAVES_PER_BLK][32][8];

    const int tid  = threadIdx.x;
    const int lane = tid & 31;
    const int wave = tid >> 5;
    const int n0   = blockIdx.x * 16;

    const int nl    = lane & 15;
    const int koff  = (lane >> 4) << 1;      // 0 or 2
    const int n     = n0 + nl;
    const int t_ln  = (n < N_COLS) ? (n / V_TOK) : -1;   // causal bound per col
    const int nld   = (n < N_COLS) ? n : (N_COLS - 1);   // clamp load address

    int t_tile = (n0 + 15) / V_TOK;
    if (t_tile > S_LEN - 1) t_tile = S_LEN - 1;
    const int sblocks = t_tile + 1;          // K-range actually needed by tile

    v8f acc = {};
    const v2f* __restrict__ xp = (const v2f*)xpack;

    for (int s = wave; s < sblocks; s += WAVES_PER_BLK) {
        const float msk = (s <= t_ln) ? 1.0f : 0.0f;   // per-lane causal mask
        const int kbase = s * W_DIM;
#pragma unroll
        for (int kk = 0; kk < W_DIM; kk += 4) {
            const int k = kbase + kk;
            v2f a = xp[(k >> 2) * 32 + lane];
            v2f b;
            b.x = weight[(size_t)(k + koff) * N_COLS + nld] * msk;
            b.y = weight[(size_t)(k + koff + 1) * N_COLS + nld] * msk;
            acc = __builtin_amdgcn_wmma_f32_16x16x4_f32(
                false, a, false, b, (short)0, acc, false, false);
        }
    }

    // Cross-wave reduction of partial C tiles through LDS.
#pragma unroll
    for (int r = 0; r < 8; ++r) lds[wave][lane][r] = acc[r];
    __syncthreads();

    if (tid < 256) {
        const int L = tid >> 3;      // lane slot 0..31
        const int r = tid & 7;       // C VGPR row 0..7
        float sum = 0.0f;
#pragma unroll
        for (int w = 0; w < WAVES_PER_BLK; ++w) sum += lds[w][L][r];
        // C layout: lane L, VGPR r  ->  M = r + (L>=16 ? 8 : 0), N = L%16
        const int m  = r + ((L >= 16) ? 8 : 0);
        const int nn = n0 + (L & 15);
        if (nn < N_COLS) {
            const int t = nn / V_TOK, v = nn % V_TOK;
            // out is (B, V, T)
            out[(size_t)m * N_COLS + (size_t)v * S_LEN + t] = sum + bias[nn];
        }
    }
}

// ---------------------------------------------------------------------------
extern "C" void kernel_launch(void* const* d_in, const int* in_sizes, int n_in,
                              void* d_out, int out_size, void* d_ws, size_t ws_size,
                              hipStream_t stream) {
    const int*   src    = (const int*)d_in[0];     // (B, S) int32
    const float* emb    = (const float*)d_in[1];   // (V, W) f32
    const float* weight = (const float*)d_in[2];   // (S, W, S, V) f32
    const float* bias   = (const float*)d_in[3];   // (S, V) f32
    float*       out    = (float*)d_out;           // (B, V, S) f32
    float*       xpack  = (float*)d_ws;            // 1.05 MB scratch

    const int total = K4_STEPS * 32;
    pack_A_kernel<<<(total + 255) / 256, 256, 0, stream>>>(src, emb, xpack);
    masked_gemm_kernel<<<N_TILES, BLK_THREADS, 0, stream>>>(xpack, weight, bias, out);
}